// HierKalmanVAE_71880572666257
// MI455X (gfx1250) — compile-verified
//
#include <hip/hip_runtime.h>

// ---------------------------------------------------------------------------
// Hierarchical Kalman filter (B=64, T=128, L=3, Z=32, ADIM=16, factor=2)
// One workgroup (4 waves) per batch chain; all state lives in LDS; every
// 32x32 / 16x32 matrix product runs on the CDNA5 f32 matrix pipe via
// V_WMMA_F32_16X16X4_F32 (fp32 kept for the Joseph-form recursion + inverse).
// Next-step operands are software-prefetched (global_prefetch_b8) to hide HBM
// latency behind the current step's WMMA work on the serial chain.
// ---------------------------------------------------------------------------

typedef __attribute__((ext_vector_type(2))) float v2f;
typedef __attribute__((ext_vector_type(8))) float v8f;

#define LD   33   // padded leading dim for 32-column LDS matrices (bank-conflict free)
#define LDK  17   // padded leading dim for 16-column LDS matrices

// ---- WMMA fragment helpers (ISA 7.12.2: 32-bit A 16x4, 32-bit C/D 16x16) ----
__device__ __forceinline__ v2f frag_a(const float* M, int ld, int r0, int k0) {
  const int l = threadIdx.x & 31;
  const int m = l & 15;
  const int kh = (l >> 4) << 1;                 // lanes 16..31 hold K=2,3
  v2f a;
  a.x = M[(r0 + m) * ld + k0 + kh];
  a.y = M[(r0 + m) * ld + k0 + kh + 1];
  return a;
}
__device__ __forceinline__ v2f frag_b(const float* M, int ld, int k0, int c0) {
  const int l = threadIdx.x & 31;
  const int n = l & 15;
  const int kh = (l >> 4) << 1;
  v2f b;
  b.x = M[(k0 + kh) * ld + c0 + n];
  b.y = M[(k0 + kh + 1) * ld + c0 + n];
  return b;
}
__device__ __forceinline__ v2f frag_bT(const float* M, int ld, int k0, int c0) {
  // B = M^T with M stored row-major [n][k]  -> transposes come for free
  const int l = threadIdx.x & 31;
  const int n = l & 15;
  const int kh = (l >> 4) << 1;
  v2f b;
  b.x = M[(c0 + n) * ld + k0 + kh];
  b.y = M[(c0 + n) * ld + k0 + kh + 1];
  return b;
}
__device__ __forceinline__ v8f tile_load(const float* M, int ld, int r0, int c0) {
  const int l = threadIdx.x & 31;
  const int n = l & 15;
  const int mh = (l >> 4) << 3;                 // lanes 16..31 hold rows 8..15
  v8f acc;
#pragma unroll
  for (int r = 0; r < 8; ++r) acc[r] = M[(r0 + mh + r) * ld + c0 + n];
  return acc;
}
__device__ __forceinline__ void tile_store(float* M, int ld, int r0, int c0, v8f acc) {
  const int l = threadIdx.x & 31;
  const int n = l & 15;
  const int mh = (l >> 4) << 3;
#pragma unroll
  for (int r = 0; r < 8; ++r) M[(r0 + mh + r) * ld + c0 + n] = acc[r];
}

// D[MxN] (+)= A[MxK] * (B | B^T).  16x16 tiles distributed across the 4 waves;
// branch is wave-uniform so EXEC stays all-ones inside the WMMA region.
__device__ __forceinline__ void wave_mm(float* Dm, int ldd,
                                        const float* Am, int lda,
                                        const float* Bm, int ldb,
                                        int M, int N, int K,
                                        bool transB, bool accum) {
  const int w = threadIdx.x >> 5;
  const int ntj = N >> 4;
  const int nt = (M >> 4) * ntj;
  if (w < nt) {
    const int i = (w / ntj) << 4;
    const int j = (w % ntj) << 4;
    v8f acc = {};
    if (accum) acc = tile_load(Dm, ldd, i, j);
#pragma unroll
    for (int k0 = 0; k0 < K; k0 += 4) {
      v2f a = frag_a(Am, lda, i, k0);
      v2f b = transB ? frag_bT(Bm, ldb, k0, j) : frag_b(Bm, ldb, k0, j);
      acc = __builtin_amdgcn_wmma_f32_16x16x4_f32(false, a, false, b,
                                                  (short)0, acc, false, false);
    }
    tile_store(Dm, ldd, i, j, acc);
  }
}

__device__ __forceinline__ void lds_load(float* dst, int ld, const float* src,
                                         int rows, int cols) {
  for (int idx = threadIdx.x; idx < rows * cols; idx += 128)
    dst[(idx / cols) * ld + (idx % cols)] = src[idx];
}
__device__ __forceinline__ void out_store(float* dst, const float* src, int ld,
                                          int rows, int cols) {
  for (int idx = threadIdx.x; idx < rows * cols; idx += 128)
    dst[idx] = src[(idx / cols) * ld + (idx % cols)];
}

// Output layout (flat concat, reference return order)
constexpr size_t O_MU_FILT  = 0;                         // [128,64,32]
constexpr size_t O_SIG_FILT = 262144;                    // [128,64,32,32]
constexpr size_t O_MU_PRED  = O_SIG_FILT + 8388608;      // [128,64,32]
constexpr size_t O_SIG_PRED = O_MU_PRED + 262144;        // [128,64,32,32]
constexpr size_t O_LAT_MU   = O_SIG_PRED + 8388608;      // [128,64,3,32]
constexpr size_t O_LAT_VAR  = O_LAT_MU + 786432;         // [128,64,3,32,32]
constexpr size_t O_S        = O_LAT_VAR + 25165824;      // [128,64,16,16]

__global__ __launch_bounds__(128)
void hier_kf_kernel(const float* __restrict__ obs, const float* __restrict__ A,
                    const float* __restrict__ C, const float* __restrict__ Dd,
                    float* __restrict__ out) {
  const int b = blockIdx.x;
  const int tid = threadIdx.x;

  __shared__ float sig2[32 * LD], sig1[32 * LD], sig0[32 * LD];
  __shared__ float As[32 * LD], Ds[32 * LD], Cs[16 * LD];
  __shared__ float T1[32 * LD], T2[32 * LD], SZ[32 * LD];
  __shared__ float Ssh[16 * LDK], Saug[16 * LD];          // Saug = [S | I] -> [S^-1]
  __shared__ float sigCt[32 * LDK], Kg[32 * LDK];
  __shared__ float mu2[32], mu1[32], mu0[32], muz[32], tv[32];
  __shared__ float rv[16];

  // ---- initial carries: mu=0.01 (levels 1,2), mu0=0, Sigma = 20*I ----
  for (int idx = tid; idx < 1024; idx += 128) {
    int r = idx >> 5, c = idx & 31;
    float e = (r == c) ? 20.0f : 0.0f;
    sig2[r * LD + c] = e; sig1[r * LD + c] = e; sig0[r * LD + c] = e;
  }
  if (tid < 32) { mu2[tid] = 0.01f; mu1[tid] = 0.01f; mu0[tid] = 0.0f; }
  __syncthreads();

  for (int t = 0; t < 128; ++t) {
    const float* Abt = A  + ((size_t)b * 128 + t) * 3072;
    const float* Dbt = Dd + ((size_t)b * 128 + t) * 3072;

    // ---- prefetch step t+1 operands (global_prefetch_b8): overlaps the HBM
    // fetch with this step's ~30 WMMA products; no LOADcnt, off the dep chain.
    if (t + 1 < 128) {
      const char* pa = (const char*)(A  + ((size_t)b * 128 + t + 1) * 3072);
      const char* pd = (const char*)(Dd + ((size_t)b * 128 + t + 1) * 3072);
      const char* pc = (const char*)(C  + ((size_t)b * 128 + t + 1) * 512);
      const int off = tid * 128;                       // 128 threads x 128B lines
      if (off < 12288) {                               // 12 KB per A/D slab
        __builtin_prefetch(pa + off, 0, 3);
        __builtin_prefetch(pd + off, 0, 3);
      }
      if (off < 2048) __builtin_prefetch(pc + off, 0, 3);
    }

    // ================= top level (l=2): every factor^2 = 4 steps ===========
    if (t > 0 && (t & 3) == 0) {
      lds_load(As, LD, Abt + 2048, 32, 32);
      __syncthreads();
      if (tid < 32) {                                   // mu2' = A*mu2 (old mu2)
        float acc = 0.f;
        for (int k = 0; k < 32; ++k) acc += As[tid * LD + k] * mu2[k];
        tv[tid] = acc;
      }
      wave_mm(T1, LD, As, LD, sig2, LD, 32, 32, 32, false, false);
      __syncthreads();
      wave_mm(T2, LD, T1, LD, As, LD, 32, 32, 32, true, false);   // A Sig A^T
      __syncthreads();
      for (int idx = tid; idx < 1024; idx += 128) {
        int r = idx >> 5, c = idx & 31;
        sig2[r * LD + c] = T2[r * LD + c] + ((r == c) ? 0.08f : 0.f);   // +O
      }
      if (tid < 32) mu2[tid] = tv[tid];
      __syncthreads();
    }

    // ================= mid level (l=1): every 2 steps ======================
    if (t > 0 && (t & 1) == 0) {
      lds_load(As, LD, Abt + 1024, 32, 32);
      lds_load(Ds, LD, Dbt + 1024, 32, 32);
      __syncthreads();
      if (tid < 32) {                                   // A*mu1 + D*mu2(post)
        float acc = 0.f;
        for (int k = 0; k < 32; ++k)
          acc += As[tid * LD + k] * mu1[k] + Ds[tid * LD + k] * mu2[k];
        tv[tid] = acc;
      }
      wave_mm(T1, LD, As, LD, sig1, LD, 32, 32, 32, false, false);
      __syncthreads();
      wave_mm(T2, LD, T1, LD, As, LD, 32, 32, 32, true, false);   // A Sig1 A^T
      __syncthreads();
      wave_mm(T1, LD, Ds, LD, sig2, LD, 32, 32, 32, false, false);
      __syncthreads();
      wave_mm(T2, LD, T1, LD, Ds, LD, 32, 32, 32, true, true);    // += D Sig2 D^T
      __syncthreads();
      for (int idx = tid; idx < 1024; idx += 128) {
        int r = idx >> 5, c = idx & 31;
        sig1[r * LD + c] = T2[r * LD + c] + ((r == c) ? 0.08f : 0.f);
      }
      if (tid < 32) mu1[tid] = tv[tid];
      __syncthreads();
    }

    // ================= latent outputs at t (l=0 is zeros) ==================
    {
      float* lm = out + O_LAT_MU  + ((size_t)t * 64 + b) * 96;
      float* lv = out + O_LAT_VAR + ((size_t)t * 64 + b) * 3072;
      if (tid < 32) {
        lm[tid]      = 0.0f;
        lm[32 + tid] = mu1[tid];
        lm[64 + tid] = mu2[tid];
      }
      for (int idx = tid; idx < 1024; idx += 128) {
        int r = idx >> 5, c = idx & 31;
        lv[idx]        = 0.0f;
        lv[1024 + idx] = sig1[r * LD + c];
        lv[2048 + idx] = sig2[r * LD + c];
      }
    }

    // ================= level 0: Kalman step ================================
    // predicted (pre-update carry) outputs
    if (tid < 32) out[O_MU_PRED + ((size_t)t * 64 + b) * 32 + tid] = mu0[tid];
    out_store(out + O_SIG_PRED + ((size_t)t * 64 + b) * 1024, sig0, LD, 32, 32);

    lds_load(Cs, LD, C + ((size_t)b * 128 + t) * 512, 16, 32);
    __syncthreads();
    if (tid < 16) {                                    // residual r = y - C mu
      float acc = obs[((size_t)b * 128 + t) * 16 + tid];
      for (int k = 0; k < 32; ++k) acc -= Cs[tid * LD + k] * mu0[k];
      rv[tid] = acc;
    }
    wave_mm(T1, LD, Cs, LD, sig0, LD, 16, 32, 32, false, false);  // C Sig
    __syncthreads();
    wave_mm(Ssh, LDK, T1, LD, Cs, LD, 16, 16, 32, true, false);   // (C Sig) C^T
    __syncthreads();
    if (tid < 16) Ssh[tid * LDK + tid] += 0.03f;                  // + R
    __syncthreads();
    out_store(out + O_S + ((size_t)t * 64 + b) * 256, Ssh, LDK, 16, 16);

    // ---- invS via column-parallel Gauss-Jordan on [S | I] (wave 0) ----
    for (int idx = tid; idx < 512; idx += 128) {
      int r = idx >> 5, c = idx & 31;
      Saug[r * LD + c] = (c < 16) ? Ssh[r * LDK + c] : ((c - 16 == r) ? 1.f : 0.f);
    }
    __syncthreads();
    if (tid < 32) {                 // single wave: LDS ops are in-order
      for (int p = 0; p < 16; ++p) {
        float inv = 1.0f / Saug[p * LD + p];
        float rp  = Saug[p * LD + tid] * inv;
        Saug[p * LD + tid] = rp;
        for (int r = 0; r < 16; ++r) {
          if (r == p) continue;
          float f = Saug[r * LD + p];       // read before this row's update
          Saug[r * LD + tid] -= f * rp;
        }
      }
    }
    __syncthreads();

    wave_mm(sigCt, LDK, sig0, LD, Cs, LD, 32, 16, 32, true, false);   // Sig C^T
    __syncthreads();
    wave_mm(Kg, LDK, sigCt, LDK, Saug + 16, LD, 32, 16, 16, false, false); // K
    __syncthreads();
    if (tid < 32) {                                   // mu_z = mu + K r
      float acc = mu0[tid];
      for (int k = 0; k < 16; ++k) acc += Kg[tid * LDK + k] * rv[k];
      muz[tid] = acc;
    }
    __syncthreads();
    if (tid < 32) out[O_MU_FILT + ((size_t)t * 64 + b) * 32 + tid] = muz[tid];

    wave_mm(T1, LD, Kg, LDK, Cs, LD, 32, 32, 16, false, false);   // K C
    __syncthreads();
    for (int idx = tid; idx < 1024; idx += 128) {                 // Iz = I - KC
      int r = idx >> 5, c = idx & 31;
      T1[r * LD + c] = ((r == c) ? 1.f : 0.f) - T1[r * LD + c];
    }
    __syncthreads();
    wave_mm(T2, LD, T1, LD, sig0, LD, 32, 32, 32, false, false);  // Iz Sig
    __syncthreads();
    wave_mm(SZ, LD, T2, LD, T1, LD, 32, 32, 32, true, false);     // .. Iz^T
    __syncthreads();
    wave_mm(T2, LD, Kg, LDK, Kg, LDK, 32, 32, 16, true, false);   // K K^T
    __syncthreads();
    for (int idx = tid; idx < 1024; idx += 128) {                 // + K R K^T
      int r = idx >> 5, c = idx & 31;
      SZ[r * LD + c] += 0.03f * T2[r * LD + c];
    }
    __syncthreads();
    out_store(out + O_SIG_FILT + ((size_t)t * 64 + b) * 1024, SZ, LD, 32, 32);

    // ---- predict with A[t+1], D[t+1] (level 0) + mid-level latent at t ----
    const int tn = (t < 127) ? t + 1 : 127;
    lds_load(As, LD, A  + ((size_t)b * 128 + tn) * 3072, 32, 32);
    lds_load(Ds, LD, Dd + ((size_t)b * 128 + tn) * 3072, 32, 32);
    __syncthreads();
    if (tid < 32) {                                   // mu0' = A mu_z + D mu1
      float acc = 0.f;
      for (int k = 0; k < 32; ++k)
        acc += As[tid * LD + k] * muz[k] + Ds[tid * LD + k] * mu1[k];
      tv[tid] = acc;
    }
    wave_mm(T1, LD, As, LD, SZ, LD, 32, 32, 32, false, false);
    __syncthreads();
    wave_mm(T2, LD, T1, LD, As, LD, 32, 32, 32, true, false);     // A Sig_z A^T
    __syncthreads();
    wave_mm(T1, LD, Ds, LD, sig1, LD, 32, 32, 32, false, false);
    __syncthreads();
    wave_mm(T2, LD, T1, LD, Ds, LD, 32, 32, 32, true, true);      // += D Sig1 D^T
    __syncthreads();
    for (int idx = tid; idx < 1024; idx += 128) {
      int r = idx >> 5, c = idx & 31;
      sig0[r * LD + c] = T2[r * LD + c] + ((r == c) ? 0.08f : 0.f);   // + Q
    }
    if (tid < 32) mu0[tid] = tv[tid];
    __syncthreads();
  }
}

extern "C" void kernel_launch(void* const* d_in, const int* in_sizes, int n_in,
                              void* d_out, int out_size, void* d_ws, size_t ws_size,
                              hipStream_t stream) {
  (void)in_sizes; (void)n_in; (void)out_size; (void)d_ws; (void)ws_size;
  const float* obs = (const float*)d_in[0];
  const float* A   = (const float*)d_in[1];
  const float* C   = (const float*)d_in[2];
  const float* Dm  = (const float*)d_in[3];
  // d_in[4] = factor (==2), baked into the kernel's update cadence.
  float* out = (float*)d_out;
  hier_kf_kernel<<<dim3(64), dim3(128), 0, stream>>>(obs, A, C, Dm, out);
}